// GAT_50603304682059
// MI455X (gfx1250) — compile-verified
//
#include <hip/hip_runtime.h>
#include <math.h>

// ---------------------------------------------------------------------------
// GAT (2-layer) for MI455X / gfx1250.
// Dense projections use V_WMMA_F32_16X16X4_F32 (fp32 in, fp32 accum) since the
// problem is bandwidth-bound (~150MB traffic @ 23.3TB/s dominates 4 GFLOP of
// GEMM) and fp32 WMMA keeps full reference precision. Edge softmax/aggregation
// use L2-resident float atomics (working set ~80MB < 192MB L2).
// ---------------------------------------------------------------------------

typedef __attribute__((ext_vector_type(2))) float v2f;
typedef __attribute__((ext_vector_type(8))) float v8f;

#define NEG_SLOPE 0.2f

__device__ __forceinline__ float lrelu(float x) {
  return x > 0.0f ? x : NEG_SLOPE * x;
}

// atomic max for signed floats: int-max for >=0, uint-min for <0.
// Works with init value -inf (0xFF800000) in both orderings.
__device__ __forceinline__ void atomicMaxF(float* addr, float val) {
  if (val >= 0.0f)
    atomicMax((int*)addr, __float_as_int(val));
  else
    atomicMin((unsigned int*)addr, __float_as_uint(val));
}

// --------------------------------- fill ------------------------------------
__global__ void fill_f32(float* __restrict__ p, float v, int n) {
  int i = blockIdx.x * blockDim.x + threadIdx.x;
  if (i < n) p[i] = v;
}

// ------------------------------ WMMA GEMM ----------------------------------
// C[M x Nc] = A[M x K] * B[K x Nc], all row-major fp32.
// Requires M % 16 == 0, Nc % 16 == 0, K % 4 == 0 (true for this problem:
// M=50000, K in {256,128}, Nc in {128,64}).
// Fragment layouts (CDNA5 ISA 7.12.2, wave32):
//   A 16x4 : lane -> row m = lane%16 ; VGPR j -> k = (lane/16)*2 + j
//   B 4x16 : lane -> col n = lane%16 ; VGPR j -> k = (lane/16)*2 + j
//   C 16x16: lane -> col n = lane%16 ; VGPR v -> row m = (lane/16)*8 + v
__global__ __launch_bounds__(256) void gemm_wmma_f32(
    const float* __restrict__ A, const float* __restrict__ B,
    float* __restrict__ C, int M, int K, int Nc) {
  const int tiles_n = Nc >> 4;
  const int tiles_m = M >> 4;
  const int wid = blockIdx.x * (blockDim.x >> 5) + (threadIdx.x >> 5);
  if (wid >= tiles_m * tiles_n) return;  // wave-uniform: EXEC stays all-ones
  const int tm = wid / tiles_n;
  const int tn = wid - tm * tiles_n;
  const int lane = threadIdx.x & 31;
  const int half = lane >> 4;
  const int lm = lane & 15;

  const float* arow = A + (size_t)(tm * 16 + lm) * K;  // this lane's A row
  const float* bcol = B + (tn * 16 + lm);              // this lane's B col

  v8f acc = {};
  for (int kk = 0; kk < K; kk += 4) {
    const int ka = kk + half * 2;
    v2f af, bf;
    af.x = arow[ka];
    af.y = arow[ka + 1];
    bf.x = bcol[(size_t)ka * Nc];
    bf.y = bcol[(size_t)(ka + 1) * Nc];
    // (neg_a, A, neg_b, B, c_mod, C, reuse_a, reuse_b)
    acc = __builtin_amdgcn_wmma_f32_16x16x4_f32(false, af, false, bf,
                                                (short)0, acc, false, false);
  }

  float* crow = C + (size_t)(tm * 16 + half * 8) * Nc + (tn * 16 + lm);
#pragma unroll
  for (int v = 0; v < 8; ++v) crow[(size_t)v * Nc] = acc[v];
}

// --------------------------- attention scores ------------------------------
// es[n,h] = sum_c h[n,h,c]*a_src[h,c] ; ed likewise with a_dst.
__global__ void attn_scores_k(const float* __restrict__ h,
                              const float* __restrict__ av_s,
                              const float* __restrict__ av_d,
                              float* __restrict__ es, float* __restrict__ ed,
                              int N, int H, int C) {
  int idx = blockIdx.x * blockDim.x + threadIdx.x;
  if (idx >= N * H) return;
  int n = idx / H, hh = idx - n * H;
  const float* hp = h + (size_t)n * H * C + (size_t)hh * C;
  float s = 0.0f, d = 0.0f;
  for (int c = 0; c < C; ++c) {
    float v = hp[c];
    s += v * av_s[hh * C + c];
    d += v * av_d[hh * C + c];
  }
  es[idx] = s;
  ed[idx] = d;
}

// ------------------------------ edge kernels -------------------------------
__device__ __forceinline__ void edge_nodes(const long long* __restrict__ A0,
                                           const long long* __restrict__ A1,
                                           int i, int E, int* s, int* t) {
  if (i < E) {
    *s = (int)A0[i];
    *t = (int)A1[i];
  } else {  // self loops appended after the E graph edges
    *s = i - E;
    *t = i - E;
  }
}

__global__ void edge_max_k(const long long* __restrict__ A0,
                           const long long* __restrict__ A1,
                           const float* __restrict__ es,
                           const float* __restrict__ ed, float* __restrict__ m,
                           int E, int Etot, int H) {
  int idx = blockIdx.x * blockDim.x + threadIdx.x;
  if (idx >= Etot * H) return;
  int i = idx / H, hh = idx - i * H;
  int s, t;
  edge_nodes(A0, A1, i, E, &s, &t);
  float e = lrelu(es[s * H + hh] + ed[t * H + hh]);
  atomicMaxF(&m[t * H + hh], e);
}

__global__ void edge_expsum_k(const long long* __restrict__ A0,
                              const long long* __restrict__ A1,
                              const float* __restrict__ es,
                              const float* __restrict__ ed,
                              const float* __restrict__ m,
                              float* __restrict__ den, int E, int Etot, int H) {
  int idx = blockIdx.x * blockDim.x + threadIdx.x;
  if (idx >= Etot * H) return;
  int i = idx / H, hh = idx - i * H;
  int s, t;
  edge_nodes(A0, A1, i, E, &s, &t);
  float e = lrelu(es[s * H + hh] + ed[t * H + hh]);
  float ex = expf(e - m[t * H + hh]);
  atomicAdd(&den[t * H + hh], ex);
}

__global__ void edge_agg_k(const long long* __restrict__ A0,
                           const long long* __restrict__ A1,
                           const float* __restrict__ es,
                           const float* __restrict__ ed,
                           const float* __restrict__ m,
                           const float* __restrict__ den,
                           const float* __restrict__ h, float* __restrict__ agg,
                           int E, int Etot, int H, int C) {
  int idx = blockIdx.x * blockDim.x + threadIdx.x;
  if (idx >= Etot * H) return;
  int i = idx / H, hh = idx - i * H;
  int s, t;
  edge_nodes(A0, A1, i, E, &s, &t);
  float e = lrelu(es[s * H + hh] + ed[t * H + hh]);
  float alpha = expf(e - m[t * H + hh]) / den[t * H + hh];
  const float* hp = h + (size_t)s * H * C + (size_t)hh * C;
  float* op = agg + (size_t)t * H * C + (size_t)hh * C;
  for (int c = 0; c < C; ++c) atomicAdd(&op[c], hp[c] * alpha);
}

// ------------------------------ epilogues ----------------------------------
__global__ void bias_elu_k(float* __restrict__ a, const float* __restrict__ b,
                           int n, int D) {
  int idx = blockIdx.x * blockDim.x + threadIdx.x;
  if (idx >= n) return;
  float v = a[idx] + b[idx % D];
  a[idx] = v > 0.0f ? v : (expf(v) - 1.0f);  // ELU(alpha=1)
}

__global__ void bias_out_k(const float* __restrict__ a,
                           const float* __restrict__ b, float* __restrict__ o,
                           int n, int D) {
  int idx = blockIdx.x * blockDim.x + threadIdx.x;
  if (idx >= n) return;
  o[idx] = a[idx] + b[idx % D];
}

// ---------------------------------------------------------------------------
extern "C" void kernel_launch(void* const* d_in, const int* in_sizes, int n_in,
                              void* d_out, int out_size, void* d_ws,
                              size_t ws_size, hipStream_t stream) {
  const float* X       = (const float*)d_in[0];
  const long long* Aix = (const long long*)d_in[1];  // int64 [2,E]
  const float* W1      = (const float*)d_in[2];
  const float* a_src1  = (const float*)d_in[3];
  const float* a_dst1  = (const float*)d_in[4];
  const float* b1      = (const float*)d_in[5];
  const float* W2      = (const float*)d_in[6];
  const float* a_src2  = (const float*)d_in[7];
  const float* a_dst2  = (const float*)d_in[8];
  const float* b2      = (const float*)d_in[9];
  float* out           = (float*)d_out;

  const int FIN = 256, H1 = 4, C1 = 32, D1 = H1 * C1 /*128*/, D2 = 64;
  const int N = in_sizes[0] / FIN;   // 50000
  const int E = in_sizes[1] / 2;     // 800000
  const int Etot = E + N;            // + self loops
  const long long* A0 = Aix;
  const long long* A1 = Aix + E;

  // ---- workspace layout (bytes) ----
  char* ws = (char*)d_ws;
  size_t off = 0;
  float* h1   = (float*)(ws + off); off += (size_t)N * D1 * 4;  // 25.6MB
  float* agg1 = (float*)(ws + off); off += (size_t)N * D1 * 4;  // 25.6MB
  float* h2   = (float*)(ws + off); off += (size_t)N * D2 * 4;  // 12.8MB
  float* agg2 = (float*)(ws + off); off += (size_t)N * D2 * 4;  // 12.8MB
  float* es1  = (float*)(ws + off); off += (size_t)N * H1 * 4;
  float* ed1  = (float*)(ws + off); off += (size_t)N * H1 * 4;
  float* m1   = (float*)(ws + off); off += (size_t)N * H1 * 4;
  float* dn1  = (float*)(ws + off); off += (size_t)N * H1 * 4;
  float* es2  = (float*)(ws + off); off += (size_t)N * 4;
  float* ed2  = (float*)(ws + off); off += (size_t)N * 4;
  float* m2   = (float*)(ws + off); off += (size_t)N * 4;
  float* dn2  = (float*)(ws + off); off += (size_t)N * 4;

  const int TB = 256;
  const float NINF = -__builtin_huge_valf();
  auto blocks = [](long long n, int tb) { return (int)((n + tb - 1) / tb); };

  // ---- reset accumulators (required every call: graph replay reuses ws) ----
  fill_f32<<<blocks((long long)N * D1, TB), TB, 0, stream>>>(agg1, 0.0f, N * D1);
  fill_f32<<<blocks((long long)N * D2, TB), TB, 0, stream>>>(agg2, 0.0f, N * D2);
  fill_f32<<<blocks((long long)N * H1, TB), TB, 0, stream>>>(m1, NINF, N * H1);
  fill_f32<<<blocks((long long)N * H1, TB), TB, 0, stream>>>(dn1, 0.0f, N * H1);
  fill_f32<<<blocks(N, TB), TB, 0, stream>>>(m2, NINF, N);
  fill_f32<<<blocks(N, TB), TB, 0, stream>>>(dn2, 0.0f, N);

  // ================= layer 1 =================
  {  // h1 = X @ W1  (50000x256 @ 256x128)
    int tiles = (N >> 4) * (D1 >> 4);
    gemm_wmma_f32<<<blocks(tiles, 8), 256, 0, stream>>>(X, W1, h1, N, FIN, D1);
  }
  attn_scores_k<<<blocks((long long)N * H1, TB), TB, 0, stream>>>(
      h1, a_src1, a_dst1, es1, ed1, N, H1, C1);
  edge_max_k<<<blocks((long long)Etot * H1, TB), TB, 0, stream>>>(
      A0, A1, es1, ed1, m1, E, Etot, H1);
  edge_expsum_k<<<blocks((long long)Etot * H1, TB), TB, 0, stream>>>(
      A0, A1, es1, ed1, m1, dn1, E, Etot, H1);
  edge_agg_k<<<blocks((long long)Etot * H1, TB), TB, 0, stream>>>(
      A0, A1, es1, ed1, m1, dn1, h1, agg1, E, Etot, H1, C1);
  bias_elu_k<<<blocks((long long)N * D1, TB), TB, 0, stream>>>(agg1, b1,
                                                              N * D1, D1);

  // ================= layer 2 =================
  {  // h2 = elu(layer1) @ W2  (50000x128 @ 128x64)
    int tiles = (N >> 4) * (D2 >> 4);
    gemm_wmma_f32<<<blocks(tiles, 8), 256, 0, stream>>>(agg1, W2, h2, N, D1, D2);
  }
  attn_scores_k<<<blocks(N, TB), TB, 0, stream>>>(h2, a_src2, a_dst2, es2, ed2,
                                                  N, 1, D2);
  edge_max_k<<<blocks(Etot, TB), TB, 0, stream>>>(A0, A1, es2, ed2, m2, E, Etot,
                                                  1);
  edge_expsum_k<<<blocks(Etot, TB), TB, 0, stream>>>(A0, A1, es2, ed2, m2, dn2,
                                                     E, Etot, 1);
  edge_agg_k<<<blocks(Etot, TB), TB, 0, stream>>>(A0, A1, es2, ed2, m2, dn2, h2,
                                                  agg2, E, Etot, 1, D2);
  bias_out_k<<<blocks((long long)N * D2, TB), TB, 0, stream>>>(agg2, b2, out,
                                                               N * D2, D2);
  (void)n_in; (void)out_size; (void)ws_size;
}